// Multi3DGrid_24223615550004
// MI455X (gfx1250) — compile-verified
//
#include <hip/hip_runtime.h>
#include <cstdint>

#ifndef __has_builtin
#define __has_builtin(x) 0
#endif

#define AS_GLOBAL __attribute__((address_space(1)))
#define AS_LDS    __attribute__((address_space(3)))

typedef int v4i __attribute__((__vector_size__(16)));

// ---------------- CDNA5 async global->LDS copy (16B per lane) ----------------
#if __has_builtin(__builtin_amdgcn_global_load_async_to_lds_b128)
#define HAVE_ASYNC_LDS 1
#else
#define HAVE_ASYNC_LDS 0
#endif

__device__ __forceinline__ void async_copy_b128(const void* gsrc, void* ldst) {
#if HAVE_ASYNC_LDS
  __builtin_amdgcn_global_load_async_to_lds_b128(
      (AS_GLOBAL v4i*)gsrc, (AS_LDS v4i*)ldst, 0, 0);
#else
  *(float4*)ldst = *(const float4*)gsrc;  // synchronous fallback (still correct)
#endif
}

__device__ __forceinline__ void wait_async_all() {
#if HAVE_ASYNC_LDS
#if __has_builtin(__builtin_amdgcn_s_wait_asynccnt)
  __builtin_amdgcn_s_wait_asynccnt(0);
#else
  asm volatile("s_wait_asynccnt 0" ::: "memory");
#endif
#endif
}

// ---------------- fast sigmoid: prefer gfx1250 v_tanh_f32 --------------------
__device__ __forceinline__ float sigmoidf_fast(float x) {
#if __has_builtin(__builtin_amdgcn_tanhf)
  return fmaf(0.5f, __builtin_amdgcn_tanhf(0.5f * x), 0.5f);
#elif __has_builtin(__builtin_amdgcn_exp2f) && __has_builtin(__builtin_amdgcn_rcpf)
  return __builtin_amdgcn_rcpf(1.0f + __builtin_amdgcn_exp2f(-1.44269504088896340736f * x));
#else
  return 1.0f / (1.0f + __expf(-x));
#endif
}

// points laid out [P]: idxs i32[P], xs f32[P,3], values f32[8,128,128,128,4], out f32[P,4]
__global__ __launch_bounds__(256) void multi_trilerp_kernel(
    const int*   __restrict__ idxs,
    const float* __restrict__ xs,
    const float* __restrict__ values,
    float*       __restrict__ out,
    int npts)
{
  __shared__ __align__(16) float s_xs[256 * 3];  // 3072 B
  __shared__ __align__(16) int   s_idx[256];     // 1024 B

  const int     t          = threadIdx.x;
  const int64_t blockStart = (int64_t)blockIdx.x * 256;
  const int64_t p          = blockStart + t;
  const int64_t rem        = (int64_t)npts - blockStart;
  const int     nb         = rem < 256 ? (int)rem : 256;

  // ---- stage this block's xs (nb*12 B) and idxs (nb*4 B) into LDS ----
  const char* gxs  = (const char*)xs   + blockStart * 12;  // 16B-aligned (3072B/block)
  const char* gidx = (const char*)idxs + blockStart * 4;   // 16B-aligned (1024B/block)
  const int xs_bytes  = nb * 12;
  const int idx_bytes = nb * 4;
  if (t * 16 < xs_bytes)  async_copy_b128(gxs  + t * 16, (char*)s_xs  + t * 16);
  if (t * 16 < idx_bytes) async_copy_b128(gidx + t * 16, (char*)s_idx + t * 16);
  wait_async_all();
  __syncthreads();

  if (p >= npts) return;

  const float x  = s_xs[3 * t + 0];
  const float y  = s_xs[3 * t + 1];
  const float z  = s_xs[3 * t + 2];
  const int   kid = s_idx[t];

  // padded-space coords: c = x*128 + 1 in [1,129)
  const float cx = x * 128.0f + 1.0f;
  const float cy = y * 128.0f + 1.0f;
  const float cz = z * 128.0f + 1.0f;
  const float fx = floorf(cx), fy = floorf(cy), fz = floorf(cz);
  const float rx = cx - fx,    ry = cy - fy,    rz = cz - fz;

  // original-grid base index in [0,127]
  int bx = (int)fx - 1;  bx = bx < 0 ? 0 : (bx > 127 ? 127 : bx);
  int by = (int)fy - 1;  by = by < 0 ? 0 : (by > 127 ? 127 : by);
  int bz = (int)fz - 1;  bz = bz < 0 ? 0 : (bz > 127 ? 127 : bz);

  const int x1 = bx + 1, y1 = by + 1, z1 = bz + 1;
  const float vx1 = (x1 < 128) ? 1.0f : 0.0f;   // upper corner in-bounds?
  const float vy1 = (y1 < 128) ? 1.0f : 0.0f;
  const float vz1 = (z1 < 128) ? 1.0f : 0.0f;
  const int x1c = (x1 < 128) ? x1 : 127;        // clamp so the load is always safe
  const int y1c = (y1 < 128) ? y1 : 127;
  const int z1c = (z1 < 128) ? z1 : 127;

  // byte offsets: element (X,Y,Z) -> ((X*128 + Y)*128 + Z)*4 floats * 4 B
  const char* vb = (const char*)values + ((size_t)kid << 25);  // kid * 128^3 * 16 B
  const size_t ox0 = (size_t)bx  << 18, ox1 = (size_t)x1c << 18;
  const size_t oy0 = (size_t)by  << 11, oy1 = (size_t)y1c << 11;
  const size_t oz0 = (size_t)bz  << 4,  oz1 = (size_t)z1c << 4;

  // 8 independent 16B gathers -> deep MLP
  const float4 g000 = *(const float4*)(vb + (ox0 + oy0 + oz0));
  const float4 g001 = *(const float4*)(vb + (ox0 + oy0 + oz1));
  const float4 g010 = *(const float4*)(vb + (ox0 + oy1 + oz0));
  const float4 g011 = *(const float4*)(vb + (ox0 + oy1 + oz1));
  const float4 g100 = *(const float4*)(vb + (ox1 + oy0 + oz0));
  const float4 g101 = *(const float4*)(vb + (ox1 + oy0 + oz1));
  const float4 g110 = *(const float4*)(vb + (ox1 + oy1 + oz0));
  const float4 g111 = *(const float4*)(vb + (ox1 + oy1 + oz1));

  // trilinear weights, zero-padding folded in via validity masks
  const float wx0 = 1.0f - rx, ax1 = rx * vx1;
  const float wy0 = 1.0f - ry, ay1 = ry * vy1;
  const float wz0 = 1.0f - rz, az1 = rz * vz1;

  float accx = 0.0f, accy = 0.0f, accz = 0.0f, accw = 0.0f;
#define ACCUM(W, G)                                   \
  do {                                                \
    const float _w = (W);                             \
    accx = fmaf(_w, sigmoidf_fast((G).x), accx);      \
    accy = fmaf(_w, sigmoidf_fast((G).y), accy);      \
    accz = fmaf(_w, sigmoidf_fast((G).z), accz);      \
    accw = fmaf(_w, sigmoidf_fast((G).w), accw);      \
  } while (0)

  ACCUM(wx0 * wy0 * wz0, g000);
  ACCUM(wx0 * wy0 * az1, g001);
  ACCUM(wx0 * ay1 * wz0, g010);
  ACCUM(wx0 * ay1 * az1, g011);
  ACCUM(ax1 * wy0 * wz0, g100);
  ACCUM(ax1 * wy0 * az1, g101);
  ACCUM(ax1 * ay1 * wz0, g110);
  ACCUM(ax1 * ay1 * az1, g111);
#undef ACCUM

  float4 result;
  result.x = accx; result.y = accy; result.z = accz; result.w = accw;
  *(float4*)((char*)out + ((size_t)p << 4)) = result;
}

extern "C" void kernel_launch(void* const* d_in, const int* in_sizes, int n_in,
                              void* d_out, int out_size, void* d_ws, size_t ws_size,
                              hipStream_t stream) {
  const int*   idxs   = (const int*)d_in[0];    // i32 [2048,1024,1]
  const float* xs     = (const float*)d_in[1];  // f32 [2048,1024,3]
  const float* values = (const float*)d_in[2];  // f32 [8,128,128,128,4]
  float*       out    = (float*)d_out;          // f32 [2048,1024,4]

  const int npts   = in_sizes[0];               // 2048*1024 points
  const int blocks = (npts + 255) / 256;
  multi_trilerp_kernel<<<blocks, 256, 0, stream>>>(idxs, xs, values, out, npts);
}